// QLayer_65481071403357
// MI455X (gfx1250) — compile-verified
//
#include <hip/hip_runtime.h>
#include <math.h>

#define NQ     12
#define DIM    4096      // 2^12
#define NPAIR  2048
#define BLK    256
#define QDEPTH 2

typedef __attribute__((ext_vector_type(2))) float v2f;
typedef __attribute__((ext_vector_type(8))) float v8f;

// ---------------------------------------------------------------------------
// Kernel 1: one workgroup (256 threads = 8 wave32) per batch element.
// Statevector lives in LDS (32KB). RX layer fused into initial product state;
// each CNOT ring fused into a single permutation pass. Emits wires 0..3 and
// the 256-wide probability colsum (for the WMMA epilogue).
// ---------------------------------------------------------------------------
__global__ __launch_bounds__(BLK) void qsim_kernel(
    const float* __restrict__ x, const float* __restrict__ params,
    float* __restrict__ out, float* __restrict__ colsum, int full_fallback)
{
    __shared__ float sre[DIM];
    __shared__ float simg[DIM];
    __shared__ float cw[NQ], sw[NQ];
    __shared__ float pc_[QDEPTH * NQ], ps_[QDEPTH * NQ];
    __shared__ float red[8][16];
    __shared__ float Rrow[16];

    const int b = blockIdx.x;
    const int t = threadIdx.x;

    if (t < NQ) {                       // per-batch RX half-angles
        float s, c;
        sincosf(0.5f * x[b * NQ + t], &s, &c);
        cw[t] = c; sw[t] = s;
    }
    if (t < QDEPTH * NQ) {              // shared RY half-angles
        float s, c;
        sincosf(0.5f * params[t], &s, &c);
        pc_[t] = c; ps_[t] = s;
    }
    __syncthreads();

    // Initial state = RX layer applied to |0..0>:
    // amp[s] = (-i)^popcount(s) * prod_w (bit_w(s) ? sin_w : cos_w)
    #pragma unroll 4
    for (int j = 0; j < 16; ++j) {
        int i = t + BLK * j;
        float mag = 1.0f;
        #pragma unroll
        for (int w = 0; w < NQ; ++w)
            mag *= ((i >> (11 - w)) & 1) ? sw[w] : cw[w];
        int ph = __popc(i) & 3;
        float re = 0.0f, im = 0.0f;
        if      (ph == 0) re =  mag;
        else if (ph == 1) im = -mag;
        else if (ph == 2) re = -mag;
        else              im =  mag;
        sre[i]  = re;
        simg[i] = im;
    }

    for (int d = 0; d < QDEPTH; ++d) {
        // ---- RY layer: 12 barriered pair-rotation passes ----
        for (int w = 0; w < NQ; ++w) {
            __syncthreads();
            const float c = pc_[d * NQ + w], s = ps_[d * NQ + w];
            const int p = 11 - w, m = 1 << p;
            #pragma unroll
            for (int kk = 0; kk < NPAIR / BLK; ++kk) {
                int k  = t + kk * BLK;
                int lo = k & (m - 1);
                int i0 = ((k - lo) << 1) | lo;
                int i1 = i0 | m;
                float ar = sre[i0], ai = simg[i0];
                float br = sre[i1], bi = simg[i1];
                sre[i0]  = c * ar - s * br;  simg[i0] = c * ai - s * bi;
                sre[i1]  = s * ar + c * br;  simg[i1] = s * ai + c * bi;
            }
        }

        // ---- CNOT ring fused into one permutation: new[s] = old[g(s)] ----
        __syncthreads();
        float rr[16], ri[16];
        #pragma unroll
        for (int j = 0; j < 16; ++j) {
            int i = t + BLK * j;
            int src = i ^ ((i & 1) << 11);          // CNOT(11,0), applied last in time
            #pragma unroll
            for (int pcb = 1; pcb <= 11; ++pcb)     // CNOT(10,11) ... CNOT(0,1)
                src ^= ((src >> pcb) & 1) << (pcb - 1);
            rr[j] = sre[src]; ri[j] = simg[src];
        }
        __syncthreads();
        #pragma unroll
        for (int j = 0; j < 16; ++j) {
            int i = t + BLK * j;
            sre[i] = rr[j]; simg[i] = ri[j];
        }
    }

    // ---- probabilities; factorized Z reduction ----
    // i = t + 256*j  ->  colsum index = t, rowsum index = j
    float csum = 0.0f;
    float rj[16];
    #pragma unroll
    for (int j = 0; j < 16; ++j) {
        int i = t + BLK * j;
        float pr = sre[i] * sre[i] + simg[i] * simg[i];
        csum += pr;
        rj[j] = pr;
    }
    if (!full_fallback)
        colsum[(size_t)b * 256 + t] = csum;

    const int wave = t >> 5, lane = t & 31;
    #pragma unroll
    for (int j = 0; j < 16; ++j)
        #pragma unroll
        for (int off = 16; off > 0; off >>= 1)
            rj[j] += __shfl_xor(rj[j], off, 32);
    if (lane == 0)
        for (int j = 0; j < 16; ++j) red[wave][j] = rj[j];
    __syncthreads();
    if (t < 16) {
        float s = 0.0f;
        for (int wv = 0; wv < 8; ++wv) s += red[wv][t];
        Rrow[t] = s;
    }
    __syncthreads();
    if (t < 4) {                       // wires 0..3 (bit 3-t of the high nibble)
        float z = 0.0f;
        for (int a = 0; a < 16; ++a)
            z += ((a >> (3 - t)) & 1) ? -Rrow[a] : Rrow[a];
        out[b * 12 + t] = z;
    }

    if (full_fallback) {               // scalar wires 4..11 (only if ws too small)
        float zw[8];
        #pragma unroll
        for (int n = 0; n < 8; ++n)
            zw[n] = ((t >> (7 - n)) & 1) ? -csum : csum;
        #pragma unroll
        for (int n = 0; n < 8; ++n)
            #pragma unroll
            for (int off = 16; off > 0; off >>= 1)
                zw[n] += __shfl_xor(zw[n], off, 32);
        __syncthreads();
        if (lane == 0)
            for (int n = 0; n < 8; ++n) red[wave][n] = zw[n];
        __syncthreads();
        if (t < 8) {
            float s = 0.0f;
            for (int wv = 0; wv < 8; ++wv) s += red[wv][t];
            out[b * 12 + 4 + t] = s;
        }
    }
}

// ---------------------------------------------------------------------------
// Kernel 2: Z8(2048x8) = C(2048x256) x signs^T(256x8) via v_wmma_f32_16x16x4.
// One wave per 16 batch rows; B fragment (+/-1 signs, zero-padded to N=16)
// generated arithmetically. Fragment layout per ISA: A/B K = reg + 2*(lane>=16),
// M/N = lane%16; D row M = vgpr + 8*(lane>=16).
// ---------------------------------------------------------------------------
__global__ __launch_bounds__(32) void zexp_wmma_kernel(
    const float* __restrict__ colsum, float* __restrict__ out)
{
    const int tile = blockIdx.x;       // 16 batch rows per wave
    const int lane = threadIdx.x;      // wave32
    const int mrow = lane & 15;
    const int half = lane >> 4;

    const float* A = colsum + (size_t)tile * 16 * 256;
    const int   nn    = mrow & 7;
    const float vmask = (mrow < 8) ? 1.0f : 0.0f;   // pad columns 8..15 with 0

    v8f acc = {};
    for (int k0 = 0; k0 < 256; k0 += 4) {
        const int ka = k0 + 2 * half;               // K = reg + 2*lane_half
        v2f a, bf;
        a.x = A[mrow * 256 + ka];
        a.y = A[mrow * 256 + ka + 1];
        // B[k][n] = sign of bit (7-n) of k  (wire 4+n has bit position 7-n)
        bf.x = vmask * (1.0f - 2.0f * (float)((ka       >> (7 - nn)) & 1));
        bf.y = vmask * (1.0f - 2.0f * (float)(((ka + 1) >> (7 - nn)) & 1));
        acc = __builtin_amdgcn_wmma_f32_16x16x4_f32(
                  false, a, false, bf, (short)0, acc, false, false);
    }

    if (mrow < 8) {
        const int w = 4 + mrow;
        #pragma unroll
        for (int v = 0; v < 8; ++v) {
            int m = v + 8 * half;                   // D row = vgpr + 8*lane_half
            out[(size_t)(tile * 16 + m) * 12 + w] = acc[v];
        }
    }
}

// ---------------------------------------------------------------------------
extern "C" void kernel_launch(void* const* d_in, const int* in_sizes, int n_in,
                              void* d_out, int out_size, void* d_ws, size_t ws_size,
                              hipStream_t stream) {
    const float* x      = (const float*)d_in[0];   // (B, 12)
    const float* params = (const float*)d_in[1];   // (24,)
    float*       out    = (float*)d_out;           // (B, 12)
    const int    B      = in_sizes[0] / NQ;        // 2048

    const size_t need = (size_t)B * 256 * sizeof(float);  // 2 MB colsum staging
    const int fallback = (ws_size < need || (B & 15) != 0) ? 1 : 0;
    float* colsum = (float*)d_ws;

    qsim_kernel<<<B, BLK, 0, stream>>>(x, params, out, colsum, fallback);
    if (!fallback)
        zexp_wmma_kernel<<<B / 16, 32, 0, stream>>>(colsum, out);
}